// Attention_45414984188085
// MI455X (gfx1250) — compile-verified
//
#include <hip/hip_runtime.h>
#include <math.h>

// CDNA5 / gfx1250: wave32, WMMA 16x16x4 fp32 tiles.
typedef __attribute__((ext_vector_type(2))) float v2f;
typedef __attribute__((ext_vector_type(8))) float v8f;

#define Bsz 8
#define Ssz 4096
#define Dsz 1024
#define ROWS 32      // S-rows per workgroup
#define TILES_PER_B (Ssz / ROWS)   // 128
#define LDSTR 1028   // 1024 + 4 pad floats: conflict-free row-strided ds_load_b64

// ---------------------------------------------------------------------------
// h_k[b][h] = sum_d k[b][d] * W2[d][h]   (tiny: 8x1024 outputs)
// ---------------------------------------------------------------------------
__global__ __launch_bounds__(256) void hk_kernel(const float* __restrict__ k,
                                                 const float* __restrict__ W2,
                                                 float* __restrict__ hk) {
  int g = blockIdx.x * 256 + threadIdx.x;   // 0 .. 8191
  int b = g >> 10;
  int h = g & 1023;
  const float* kb = k + b * Dsz;
  float s = 0.f;
  for (int d = 0; d < Dsz; ++d) s = fmaf(kb[d], W2[d * Dsz + h], s);
  hk[g] = s;
}

// ---------------------------------------------------------------------------
// One N-half of a 32x1024 = (32x1024 LDS) @ (1024x1024 global) fp32 GEMM.
// Wave owns 32 rows x 64 cols = 2 row-tiles x 4 col-tiles; every B fragment
// feeds TWO wmma ops (A row-tile 0 and 1) -> 1 b32-load-pair per 2 WMMAs.
// A frag (16x4): lanes 0-15 M=0..15 K={k0,k0+1}; lanes 16-31 K={k0+2,k0+3}.
// B frag (4x16): lanes 0-15 K={k0,k0+1} N=n+lane; lanes 16-31 K={k0+2,k0+3}.
// ---------------------------------------------------------------------------
__device__ inline void gemm_half(const float* __restrict__ Wmat,
                                 const float* __restrict__ Abuf,
                                 int colbase, int lane, v8f acc[2][4]) {
  const int khalf = (lane >> 4) << 1;          // 0 or 2
  const int r0    = lane & 15;
  const int nlane = colbase + (lane & 15);
  for (int k0 = 0; k0 < Dsz; k0 += 4) {
    // Speculative prefetch of a future K-step's row segment (safe past end).
    __builtin_prefetch(&Wmat[(k0 + 16) * Dsz + colbase], 0, 3);
    const int krow = k0 + khalf;
    float2 t0 = *reinterpret_cast<const float2*>(&Abuf[r0 * LDSTR + k0 + khalf]);
    float2 t1 = *reinterpret_cast<const float2*>(&Abuf[(r0 + 16) * LDSTR + k0 + khalf]);
    v2f a0; a0.x = t0.x; a0.y = t0.y;
    v2f a1; a1.x = t1.x; a1.y = t1.y;
#pragma unroll
    for (int ct = 0; ct < 4; ++ct) {
      const int ncol = nlane + ct * 16;
      v2f bv;
      bv.x = Wmat[krow * Dsz + ncol];
      bv.y = Wmat[(krow + 1) * Dsz + ncol];
      acc[0][ct] = __builtin_amdgcn_wmma_f32_16x16x4_f32(
          false, a0, false, bv, (short)0, acc[0][ct], false, false);
      acc[1][ct] = __builtin_amdgcn_wmma_f32_16x16x4_f32(
          false, a1, false, bv, (short)0, acc[1][ct], false, false);
    }
  }
}

// ---------------------------------------------------------------------------
// One workgroup = one (batch, 32-row S tile). 256 threads = 8 wave32.
// LDS: qs (q tile, becomes scores/probs buffer after GEMM1) + hs (tanh acts).
// ---------------------------------------------------------------------------
__global__ __launch_bounds__(256) void attn_main(const float* __restrict__ q,
                                                 const float* __restrict__ W1,
                                                 const float* __restrict__ Vm,
                                                 const float* __restrict__ hk,
                                                 float* __restrict__ weights_out,
                                                 float* __restrict__ partials) {
  __shared__ float qs[ROWS * LDSTR];   // q tile -> later scores -> probs
  __shared__ float hs[ROWS * LDSTR];   // tanh(h_q + h_k)

  const int tile = blockIdx.x;              // 0 .. B*TILES_PER_B-1
  const int b    = tile >> 7;               // 128 tiles per batch
  const int s0   = (tile & 127) << 5;
  const int tid  = threadIdx.x;
  const int lane = tid & 31;
  const int w    = tid >> 5;

  const float* qg = q + ((size_t)b * Ssz + s0) * Dsz;   // this tile's q rows

  // ---- stage q tile: 32 x 1024 f32 (float4 global -> b128 LDS)
  {
    const float4* q4 = reinterpret_cast<const float4*>(qg);
#pragma unroll
    for (int r = 0; r < ROWS; ++r) {
      float4 v4 = q4[r * (Dsz / 4) + tid];
      float* dst = &qs[r * LDSTR + tid * 4];
      dst[0] = v4.x; dst[1] = v4.y; dst[2] = v4.z; dst[3] = v4.w;
    }
  }
  __syncthreads();

  const int rbase = (lane >> 4) << 3;    // C/D layout: lanes 16-31 hold M=r+8

  // ---- GEMM1: h = q_tile @ W1 ; epilogue tanh(h + h_k) -> hs
  for (int half = 0; half < 2; ++half) {
    const int colbase = half * 512 + w * 64;
    v8f acc[2][4] = {};
    gemm_half(W1, qs, colbase, lane, acc);
#pragma unroll
    for (int rt = 0; rt < 2; ++rt)
#pragma unroll
      for (int ct = 0; ct < 4; ++ct) {
        const int col = colbase + ct * 16 + (lane & 15);
        const float hkv = hk[b * Dsz + col];
#pragma unroll
        for (int r = 0; r < 8; ++r)
          hs[(rt * 16 + rbase + r) * LDSTR + col] = tanhf(acc[rt][ct][r] + hkv);
      }
  }
  __syncthreads();   // hs complete; qs no longer needed (q re-read from L2 later)

  // ---- GEMM2: scores = hs @ V ; epilogue -> qs (reused as score buffer)
  for (int half = 0; half < 2; ++half) {
    const int colbase = half * 512 + w * 64;
    v8f acc[2][4] = {};
    gemm_half(Vm, hs, colbase, lane, acc);
#pragma unroll
    for (int rt = 0; rt < 2; ++rt)
#pragma unroll
      for (int ct = 0; ct < 4; ++ct) {
        const int col = colbase + ct * 16 + (lane & 15);
#pragma unroll
        for (int r = 0; r < 8; ++r)
          qs[(rt * 16 + rbase + r) * LDSTR + col] = acc[rt][ct][r];
      }
  }
  __syncthreads();

  // ---- softmax over feature dim (1024): wave w owns rows 4w .. 4w+3
#pragma unroll
  for (int rr = 0; rr < 4; ++rr) {
    const int row = 4 * w + rr;
    float* srow = &qs[row * LDSTR];
    float m = -INFINITY;
    for (int c = lane; c < Dsz; c += 32) m = fmaxf(m, srow[c]);
#pragma unroll
    for (int off = 16; off; off >>= 1) m = fmaxf(m, __shfl_xor(m, off, 32));
    float ssum = 0.f;
    for (int c = lane; c < Dsz; c += 32) {
      float e = __expf(srow[c] - m);
      srow[c] = e;
      ssum += e;
    }
#pragma unroll
    for (int off = 16; off; off >>= 1) ssum += __shfl_xor(ssum, off, 32);
    const float inv = 1.f / ssum;
    const size_t ob = ((size_t)b * Ssz + s0 + row) * Dsz;
    for (int c = lane; c < Dsz; c += 32) {
      float p = srow[c] * inv;
      srow[c] = p;                  // keep probs for the weights*q partial
      weights_out[ob + c] = p;
    }
  }
  __syncthreads();

  // ---- deterministic per-tile partial: values[b][d] += sum_r p[r][d]*q[r][d]
  //      (q re-read from global; tile is L2-hot)
  for (int d = tid; d < Dsz; d += 256) {
    float s = 0.f;
#pragma unroll
    for (int r = 0; r < ROWS; ++r)
      s = fmaf(qs[r * LDSTR + d], qg[(size_t)r * Dsz + d], s);
    partials[(size_t)tile * Dsz + d] = s;
  }
}

// ---------------------------------------------------------------------------
// values[b][d] = fixed-order sum over the 128 S-tiles of that batch
// ---------------------------------------------------------------------------
__global__ __launch_bounds__(256) void reduce_kernel(const float* __restrict__ partials,
                                                     float* __restrict__ values) {
  int g = blockIdx.x * 256 + threadIdx.x;   // 0 .. 8191  (b*1024 + d)
  int b = g >> 10;
  int d = g & 1023;
  const float* p = partials + (size_t)b * TILES_PER_B * Dsz + d;
  float s = 0.f;
  for (int t = 0; t < TILES_PER_B; ++t) s += p[(size_t)t * Dsz];
  values[g] = s;
}

extern "C" void kernel_launch(void* const* d_in, const int* in_sizes, int n_in,
                              void* d_out, int out_size, void* d_ws, size_t ws_size,
                              hipStream_t stream) {
  (void)in_sizes; (void)n_in; (void)out_size; (void)ws_size;
  const float* q  = (const float*)d_in[0];
  const float* k  = (const float*)d_in[1];
  const float* W1 = (const float*)d_in[2];
  const float* W2 = (const float*)d_in[3];
  const float* Vm = (const float*)d_in[4];

  float* out     = (float*)d_out;
  float* values  = out;                       // [8,1024]
  float* weights = out + Bsz * Dsz;           // [8,4096,1024]

  float* partials = (float*)d_ws;             // 8*128*1024 f32 = 4 MB
  float* hk       = partials + (size_t)Bsz * TILES_PER_B * Dsz;  // + 8*1024 f32

  hk_kernel<<<(Bsz * Dsz) / 256, 256, 0, stream>>>(k, W2, hk);
  attn_main<<<Bsz * TILES_PER_B, 256, 0, stream>>>(q, W1, Vm, hk, weights, partials);
  reduce_kernel<<<(Bsz * Dsz) / 256, 256, 0, stream>>>(partials, values);
}